// GCNNet_34746285424664
// MI455X (gfx1250) — compile-verified
//
#include <hip/hip_runtime.h>
#include <cstdint>
#include <cstddef>

typedef float v2f __attribute__((ext_vector_type(2)));
typedef float v8f __attribute__((ext_vector_type(8)));

#define F_IN   128
#define HID    64
#define NOUT   40
#define NOUTP  48     // NOUT padded to 3x16 WMMA tiles
// LDS k-pair-interleaved layouts: row stride in floats, chosen >= 2*Ncols and
// ≡ 32 (mod 64) so the two half-wave groups of a ds_load_b64 use disjoint banks.
#define PSTR1  160    // GEMM1: 2*64 used + 32 pad   (64 pair-rows -> 40 KB)
#define PSTR2  96     // GEMM2: 2*48 used            (32 pair-rows -> 12 KB)

static __device__ __forceinline__ v8f wmma_f32(v2f a, v2f b, v8f c) {
  // V_WMMA_F32_16X16X4_F32: D = A(16x4) * B(4x16) + C(16x16), exact fp32
  return __builtin_amdgcn_wmma_f32_16x16x4_f32(false, a, false, b, (short)0, c,
                                               false, false);
}

// ---------------- degree / normalization ----------------
__global__ void k_zero_u32(unsigned* __restrict__ p, int n) {
  int i = blockIdx.x * blockDim.x + threadIdx.x;
  if (i < n) p[i] = 0u;
}

__global__ void k_count(const int* __restrict__ col, unsigned* __restrict__ cnt,
                        int E_) {
  int e = blockIdx.x * blockDim.x + threadIdx.x;
  if (e < E_) atomicAdd(&cnt[col[e]], 1u);
}

// in place: u32 count -> float dinv = rsqrt(count + 1)   (+1 = self loop)
__global__ void k_dinv(float* __restrict__ dinv, int n) {
  int i = blockIdx.x * blockDim.x + threadIdx.x;
  if (i < n) {
    unsigned c = ((const unsigned*)dinv)[i];
    dinv[i] = rsqrtf((float)(c + 1u));
  }
}

__global__ void k_edgew(const int* __restrict__ row, const int* __restrict__ col,
                        const float* __restrict__ dinv, float* __restrict__ w,
                        int E_) {
  int e = blockIdx.x * blockDim.x + threadIdx.x;
  if (e < E_) w[e] = dinv[row[e]] * dinv[col[e]];
}

// ---------------- GEMM1: h1 = x @ W1  (100000x128 @ 128x64) ----------------
// 1 wave -> 16x64 tile (4 accumulators). W1 in LDS, k-pair interleaved so each
// B fragment is a single aligned ds_load_b64 (no register reassembly moves).
__global__ __launch_bounds__(128) void k_gemm1(const float* __restrict__ x,
                                               const float* __restrict__ W,
                                               float* __restrict__ h, int nrows) {
  __shared__ float sW[(F_IN / 2) * PSTR1];
  for (int i = threadIdx.x; i < F_IN * HID; i += 128) {
    int kk = i >> 6, nn = i & 63;                 // W is [128][64] row-major
    sW[(kk >> 1) * PSTR1 + nn * 2 + (kk & 1)] = W[i];
  }
  __syncthreads();

  const int lane = threadIdx.x & 31;
  const int wave = threadIdx.x >> 5;
  const int row0 = (blockIdx.x * 4 + wave) * 16;
  if (row0 >= nrows) return;               // wave-uniform: EXEC stays all-ones

  const int m  = lane & 15;                // A: row within tile / B,C: column
  const int hi = lane >> 4;                // 0: K=k,k+1   1: K=k+2,k+3
  const float* __restrict__ xr = x + (size_t)(row0 + m) * F_IN;

  v8f a0 = {}, a1 = {}, a2 = {}, a3 = {};
  for (int k = 0; k < F_IN; k += 4) {
    v2f a = *(const v2f*)(xr + k + hi * 2);
    const float* wp = &sW[((k >> 1) + hi) * PSTR1];
    v2f b0 = *(const v2f*)(wp + 2 * m);
    v2f b1 = *(const v2f*)(wp + 2 * (16 + m));
    v2f b2 = *(const v2f*)(wp + 2 * (32 + m));
    v2f b3 = *(const v2f*)(wp + 2 * (48 + m));
    a0 = wmma_f32(a, b0, a0);
    a1 = wmma_f32(a, b1, a1);
    a2 = wmma_f32(a, b2, a2);
    a3 = wmma_f32(a, b3, a3);
  }
  const int mb = hi << 3;                  // C/D: rows 0..7 / 8..15
  float* hp = h + (size_t)(row0 + mb) * HID + m;
  for (int i = 0; i < 8; ++i) {
    hp[(size_t)i * HID +  0] = a0[i];
    hp[(size_t)i * HID + 16] = a1[i];
    hp[(size_t)i * HID + 32] = a2[i];
    hp[(size_t)i * HID + 48] = a3[i];
  }
}

// ---------------- layer-1 aggregation ----------------
// self-loop + bias: agg1 = h1 * dinv^2 + b1
__global__ void k_init_agg1(const float* __restrict__ h1,
                            const float* __restrict__ dinv,
                            const float* __restrict__ b1,
                            float* __restrict__ agg, int total) {
  int i = blockIdx.x * blockDim.x + threadIdx.x;
  if (i >= total) return;
  float d = dinv[i >> 6];
  agg[i] = h1[i] * d * d + b1[i & 63];
}

// 32 threads per edge, float2 per lane: coalesced b64 gather + f32 atomics
__global__ void k_scatter1(const int* __restrict__ row, const int* __restrict__ col,
                           const float* __restrict__ w, const float* __restrict__ h,
                           float* __restrict__ agg, int E_) {
  long long t = (long long)blockIdx.x * blockDim.x + threadIdx.x;
  int e = (int)(t >> 5);
  if (e >= E_) return;
  int f = ((int)t & 31) * 2;
  int r = row[e], c = col[e];
  float we = w[e];
  v2f hv = *(const v2f*)(h + (size_t)r * HID + f);
  float* dst = agg + (size_t)c * HID + f;
  atomicAdd(dst,     hv.x * we);
  atomicAdd(dst + 1, hv.y * we);
}

// ---------------- GEMM2: h2 = relu(agg1) @ W2  (100000x64 @ 64x40) ----------
__global__ __launch_bounds__(128) void k_gemm2(const float* __restrict__ agg,
                                               const float* __restrict__ W2,
                                               float* __restrict__ h2, int nrows) {
  __shared__ float sW[(HID / 2) * PSTR2];
  for (int i = threadIdx.x; i < HID * NOUTP; i += 128) {
    int kk = i / NOUTP, nn = i % NOUTP;
    float v = (nn < NOUT) ? W2[kk * NOUT + nn] : 0.0f;
    sW[(kk >> 1) * PSTR2 + nn * 2 + (kk & 1)] = v;
  }
  __syncthreads();

  const int lane = threadIdx.x & 31;
  const int wave = threadIdx.x >> 5;
  const int row0 = (blockIdx.x * 4 + wave) * 16;
  if (row0 >= nrows) return;

  const int m  = lane & 15;
  const int hi = lane >> 4;
  const float* __restrict__ ar = agg + (size_t)(row0 + m) * HID;

  v8f a0 = {}, a1 = {}, a2 = {};
  for (int k = 0; k < HID; k += 4) {
    v2f av = *(const v2f*)(ar + k + hi * 2);
    v2f a;
    a.x = fmaxf(av.x, 0.0f);               // fused ReLU
    a.y = fmaxf(av.y, 0.0f);
    const float* wp = &sW[((k >> 1) + hi) * PSTR2];
    v2f b0 = *(const v2f*)(wp + 2 * m);
    v2f b1 = *(const v2f*)(wp + 2 * (16 + m));
    v2f b2 = *(const v2f*)(wp + 2 * (32 + m));
    a0 = wmma_f32(a, b0, a0);
    a1 = wmma_f32(a, b1, a1);
    a2 = wmma_f32(a, b2, a2);
  }
  const int mb = hi << 3;
  float* hp = h2 + (size_t)(row0 + mb) * NOUT;
  for (int i = 0; i < 8; ++i) {
    hp[(size_t)i * NOUT + m]      = a0[i];
    hp[(size_t)i * NOUT + 16 + m] = a1[i];
    if (m < NOUT - 32)                      // cols 32..39 only
      hp[(size_t)i * NOUT + 32 + m] = a2[i];
  }
}

// ---------------- layer-2 aggregation (into d_out) ----------------
__global__ void k_init_out(const float* __restrict__ h2,
                           const float* __restrict__ dinv,
                           const float* __restrict__ b2,
                           float* __restrict__ out, int total) {
  int i = blockIdx.x * blockDim.x + threadIdx.x;
  if (i >= total) return;
  int node = i / NOUT, f = i % NOUT;
  float d = dinv[node];
  out[i] = h2[i] * d * d + b2[f];
}

// 32 threads per edge; lanes 0..19 each handle a float2 of the 40 features
__global__ void k_scatter2(const int* __restrict__ row, const int* __restrict__ col,
                           const float* __restrict__ w, const float* __restrict__ h2,
                           float* __restrict__ out, int E_) {
  long long t = (long long)blockIdx.x * blockDim.x + threadIdx.x;
  int e = (int)(t >> 5);
  if (e >= E_) return;
  int l = (int)t & 31;
  if (l >= NOUT / 2) return;
  int f = l * 2;
  int r = row[e], c = col[e];
  float we = w[e];
  v2f hv = *(const v2f*)(h2 + (size_t)r * NOUT + f);
  float* dst = out + (size_t)c * NOUT + f;
  atomicAdd(dst,     hv.x * we);
  atomicAdd(dst + 1, hv.y * we);
}

// ---------------- log_softmax, one wave32 per node, in place --------------
__global__ __launch_bounds__(256) void k_lsm(float* __restrict__ out, int n) {
  const int lane = threadIdx.x & 31;
  const int node = blockIdx.x * 8 + (threadIdx.x >> 5);
  if (node >= n) return;                    // wave-uniform
  float* p = out + (size_t)node * NOUT;
  float v0 = p[lane];
  float v1 = (lane < NOUT - 32) ? p[32 + lane] : -INFINITY;
  float mx = fmaxf(v0, v1);
  for (int s = 16; s >= 1; s >>= 1) mx = fmaxf(mx, __shfl_xor(mx, s, 32));
  float sm = __expf(v0 - mx) + ((lane < NOUT - 32) ? __expf(v1 - mx) : 0.0f);
  for (int s = 16; s >= 1; s >>= 1) sm += __shfl_xor(sm, s, 32);
  float lse = mx + __logf(sm);
  p[lane] = v0 - lse;
  if (lane < NOUT - 32) p[32 + lane] = v1 - lse;
}

// ---------------- host side ----------------
static inline size_t align_up(size_t v) { return (v + 255) & ~(size_t)255; }

extern "C" void kernel_launch(void* const* d_in, const int* in_sizes, int n_in,
                              void* d_out, int out_size, void* d_ws, size_t ws_size,
                              hipStream_t stream) {
  const float* x  = (const float*)d_in[0];
  const int*   ei = (const int*)d_in[1];
  const float* W1 = (const float*)d_in[2];
  const float* b1 = (const float*)d_in[3];
  const float* W2 = (const float*)d_in[4];
  const float* b2 = (const float*)d_in[5];
  const int N_ = in_sizes[0] / F_IN;   // 100000
  const int E_ = in_sizes[1] / 2;      // 1600000
  const int* row = ei;                 // edge_index[0]
  const int* col = ei + E_;            // edge_index[1]
  float* out = (float*)d_out;

  // workspace layout
  char* ws = (char*)d_ws;
  size_t off = 0;
  float* dinv = (float*)(ws + off); off += align_up((size_t)N_ * 4);
  float* ew   = (float*)(ws + off); off += align_up((size_t)E_ * 4);
  float* h1   = (float*)(ws + off); off += align_up((size_t)N_ * HID * 4);
  float* agg1 = (float*)(ws + off); off += align_up((size_t)N_ * HID * 4);
  float* h2   = (float*)(ws + off); off += align_up((size_t)N_ * NOUT * 4);
  (void)ws_size; (void)n_in; (void)out_size;

  const int T = 256;
  const int gbN   = (N_ + T - 1) / T;
  const int gbE   = (E_ + T - 1) / T;
  const int tiles = (N_ + 15) / 16;
  const int gbMM  = (tiles + 3) / 4;
  const int gbA1  = ((N_ * HID) + T - 1) / T;
  const int gbS   = (int)(((long long)E_ * 32 + T - 1) / T);
  const int gbO   = ((N_ * NOUT) + T - 1) / T;
  const int gbL   = (N_ + 7) / 8;

  // normalization
  k_zero_u32 <<<gbN, T, 0, stream>>>((unsigned*)dinv, N_);
  k_count    <<<gbE, T, 0, stream>>>(col, (unsigned*)dinv, E_);
  k_dinv     <<<gbN, T, 0, stream>>>(dinv, N_);
  k_edgew    <<<gbE, T, 0, stream>>>(row, col, dinv, ew, E_);

  // layer 1
  k_gemm1    <<<gbMM, 128, 0, stream>>>(x, W1, h1, N_);
  k_init_agg1<<<gbA1, T, 0, stream>>>(h1, dinv, b1, agg1, N_ * HID);
  k_scatter1 <<<gbS, T, 0, stream>>>(row, col, ew, h1, agg1, E_);

  // layer 2 (ReLU fused into GEMM2 A-load)
  k_gemm2    <<<gbMM, 128, 0, stream>>>(agg1, W2, h2, N_);
  k_init_out <<<gbO, T, 0, stream>>>(h2, dinv, b2, out, N_ * NOUT);
  k_scatter2 <<<gbS, T, 0, stream>>>(row, col, ew, h2, out, E_);

  // log_softmax
  k_lsm      <<<gbL, T, 0, stream>>>(out, N_);
}